// CropResize_8186207666329
// MI455X (gfx1250) — compile-verified
//
#include <hip/hip_runtime.h>
#include <cstdint>
#include <cstddef>

#ifndef __has_builtin
#define __has_builtin(x) 0
#endif

// Problem constants (from reference)
#define KBOX   1024
#define CHN    256
#define IH     200
#define IW     200
#define OHH    14
#define OWW    14
#define NPIX   (OHH * OWW)        // 196
#define CCHUNK 64                 // channels per block
#define NCHUNK (CHN / CCHUNK)     // 4
#define TPB    224                // 7 full wave32s
#define OUTCH  (CCHUNK * NPIX)    // 12544 floats staged in LDS
#define NV4    (OUTCH / 4)        // 3136 float4s
#define V4ITER (NV4 / TPB)        // 14 exactly

// Typed, address-space-qualified pointers for the async builtin
// (clang's builtin signature: (v4i global*, v4i lds*, imm offset, imm cpol))
typedef int v4i __attribute__((ext_vector_type(4)));
typedef __attribute__((address_space(1))) v4i* gv4i_p;
typedef __attribute__((address_space(3))) v4i* lv4i_p;

// Dense 16B LDS->global async store (ASYNCcnt-tracked, CDNA5 path)
__device__ __forceinline__ void async_store_b128(void* gaddr, void* laddr) {
#if __has_builtin(__builtin_amdgcn_global_store_async_from_lds_b128)
  __builtin_amdgcn_global_store_async_from_lds_b128(
      (gv4i_p)(uintptr_t)gaddr,
      (lv4i_p)(uint32_t)(uintptr_t)laddr,   // low 32 bits of generic LDS ptr = LDS offset
      0, 0);
#else
  uint64_t ga = (uint64_t)(uintptr_t)gaddr;
  uint32_t la = (uint32_t)(uintptr_t)laddr;
  asm volatile("global_store_async_from_lds_b128 %0, %1, off"
               :: "v"(ga), "v"(la) : "memory");
#endif
}

__device__ __forceinline__ void wait_async0() {
#if __has_builtin(__builtin_amdgcn_s_wait_asynccnt)
  __builtin_amdgcn_s_wait_asynccnt(0);
#else
  asm volatile("s_wait_asynccnt 0" ::: "memory");
#endif
}

__global__ __launch_bounds__(TPB)
void crop_resize_kernel(const float* __restrict__ img,
                        const float* __restrict__ boxes,
                        float* __restrict__ out) {
  __shared__ float stage[OUTCH];   // 50176 B

  const int k     = blockIdx.x;    // box
  const int chunk = blockIdx.y;    // channel chunk
  const int t     = threadIdx.x;

  if (t < NPIX) {
    const int oy = t / OWW;
    const int ox = t % OWW;

    // Box (SCALE == 1.0, so no divide)
    const float xmin = boxes[4 * k + 0];
    const float ymin = boxes[4 * k + 1];
    const float xmax = boxes[4 * k + 2];
    const float ymax = boxes[4 * k + 3];

    // Replicate reference arithmetic chain in f32
    const float a11 = (xmax - xmin) / 14.0f;
    const float a22 = (ymax - ymin) / 14.0f;
    const float xi  = a11 * (float)ox + xmin;
    const float yi  = a22 * (float)oy + ymin;
    const float gx  = (xi - 100.0f) / 200.0f * 2.0f;
    const float gy  = (yi - 100.0f) / 200.0f * 2.0f;
    const float x   = ((gx + 1.0f) * 200.0f - 1.0f) / 2.0f;
    const float y   = ((gy + 1.0f) * 200.0f - 1.0f) / 2.0f;

    const float x0f = floorf(x);
    const float y0f = floorf(y);
    const float wx1 = x - x0f;
    const float wy1 = y - y0f;
    const float wx0 = 1.0f - wx1;
    const float wy0 = 1.0f - wy1;

    const int x0 = (int)x0f, y0 = (int)y0f;
    const int x1 = x0 + 1,   y1 = y0 + 1;

    const int cx0 = x0 < 0 ? 0 : (x0 > IW - 1 ? IW - 1 : x0);
    const int cx1 = x1 < 0 ? 0 : (x1 > IW - 1 ? IW - 1 : x1);
    const int cy0 = y0 < 0 ? 0 : (y0 > IH - 1 ? IH - 1 : y0);
    const int cy1 = y1 < 0 ? 0 : (y1 > IH - 1 ? IH - 1 : y1);

    const float vx0 = (x0 >= 0 && x0 < IW) ? 1.0f : 0.0f;
    const float vx1 = (x1 >= 0 && x1 < IW) ? 1.0f : 0.0f;
    const float vy0 = (y0 >= 0 && y0 < IH) ? 1.0f : 0.0f;
    const float vy1 = (y1 >= 0 && y1 < IH) ? 1.0f : 0.0f;

    const int i00 = cy0 * IW + cx0;
    const int i01 = cy0 * IW + cx1;
    const int i10 = cy1 * IW + cx0;
    const int i11 = cy1 * IW + cx1;

    // validity folded into weights (reference: v * valid * weight)
    const float w00 = wy0 * wx0 * vy0 * vx0;
    const float w01 = wy0 * wx1 * vy0 * vx1;
    const float w10 = wy1 * wx0 * vy1 * vx0;
    const float w11 = wy1 * wx1 * vy1 * vx1;

    const float* base = img + (size_t)(chunk * CCHUNK) * (IH * IW);
#pragma unroll 4
    for (int c = 0; c < CCHUNK; ++c) {
      const float* p = base + (size_t)c * (IH * IW);
      const float v = w00 * p[i00] + w01 * p[i01] + w10 * p[i10] + w11 * p[i11];
      stage[c * NPIX + t] = v;   // consecutive lanes -> consecutive banks
    }
  }

  __syncthreads();  // make all LDS writes visible before async drain

  // Drain the 50KB tile with dense b128 async stores: all 224 lanes,
  // 14 iterations each (3136 float4s), perfectly coalesced 16B stores.
  float* outk = out + (size_t)k * (CHN * NPIX) + (size_t)chunk * OUTCH;
#pragma unroll
  for (int it = 0; it < V4ITER; ++it) {
    const int e = it * TPB + t;           // float4 index
    async_store_b128((void*)(outk + 4 * e), (void*)(stage + 4 * e));
  }
  wait_async0();
}

extern "C" void kernel_launch(void* const* d_in, const int* in_sizes, int n_in,
                              void* d_out, int out_size, void* d_ws, size_t ws_size,
                              hipStream_t stream) {
  (void)in_sizes; (void)n_in; (void)out_size; (void)d_ws; (void)ws_size;
  const float* data  = (const float*)d_in[0];   // [1,256,200,200] f32
  const float* boxes = (const float*)d_in[1];   // [1024,4] f32
  float* out = (float*)d_out;                   // [1024,256,14,14] f32

  dim3 grid(KBOX, NCHUNK);
  crop_resize_kernel<<<grid, TPB, 0, stream>>>(data, boxes, out);
}